// QLSTM_65481071397858
// MI455X (gfx1250) — compile-verified
//
#include <hip/hip_runtime.h>

typedef __attribute__((ext_vector_type(2))) float v2f;
typedef __attribute__((ext_vector_type(4))) float v4f;
typedef __attribute__((ext_vector_type(8))) float v8f;

constexpr int Tn = 512;   // seq_len
constexpr int Bn = 256;   // batch
constexpr int Dn = 512;   // input dim
constexpr int Hn = 512;   // hidden dim
constexpr int NP = 16;    // 4 gates x 4 params

__device__ __forceinline__ float fast_sigmoid(float x) {
    return __builtin_amdgcn_rcpf(1.0f + __expf(-x));
}
__device__ __forceinline__ float fast_tanh(float x) {
    float t = __expf(-2.0f * x);
    return (1.0f - t) * __builtin_amdgcn_rcpf(1.0f + t);
}

// ---------------------------------------------------------------------------
// Kernel 0: pack weights.
//   Bmat[d][p] = Wgate[param][d]        (k-major B operand for WMMA), d < D
//   b16[p]     = bias
//   ws16[p]    = sum_j Wgate[param][D+j] (hidden part collapses to a rowsum
//                because h[b,:] is constant across the hidden dimension)
// ---------------------------------------------------------------------------
__global__ void qlstm_pack(const float* __restrict__ Wf, const float* __restrict__ bf,
                           const float* __restrict__ Wi, const float* __restrict__ bi,
                           const float* __restrict__ Wu, const float* __restrict__ bu,
                           const float* __restrict__ Wo, const float* __restrict__ bo,
                           float* __restrict__ Bmat, float* __restrict__ b16,
                           float* __restrict__ ws16) {
    const float* Wg[4] = {Wf, Wi, Wu, Wo};
    const float* bg[4] = {bf, bi, bu, bo};
    int tid = threadIdx.x;

    if (tid < Dn) {
        #pragma unroll
        for (int p = 0; p < NP; ++p) {
            int g = p >> 2, prm = p & 3;
            Bmat[tid * NP + p] = Wg[g][prm * (Dn + Hn) + tid];
        }
    }
    if (tid < NP) b16[tid] = bg[tid >> 2][tid & 3];

    int wave = tid >> 5, lane = tid & 31;
    if (wave < NP) {
        int g = wave >> 2, prm = wave & 3;
        const float* wrow = Wg[g] + prm * (Dn + Hn) + Dn;
        float s = 0.0f;
        for (int j = lane; j < Hn; j += 32) s += wrow[j];
        #pragma unroll
        for (int off = 16; off; off >>= 1) s += __shfl_xor(s, off, 32);
        if (lane == 0) ws16[wave] = s;
    }
}

// ---------------------------------------------------------------------------
// Kernel 1: Zx[row][p] = sum_d X[row][d] * W16[p][d] + b16[p]
// rows = T*B = 131072, K = D = 512, N = 16.
// One wave computes a 16x16 output tile with 128 V_WMMA_F32_16X16X4_F32 ops.
// A (16x4 f32): lanes 0-15 hold M=lane, K={0,1}; lanes 16-31 hold M=lane-16,
// K={2,3} (reg0=even K, reg1=odd K). B (4x16): mirrored (K across half-waves,
// N=lane&15). C/D: VGPR r -> (M=r | r+8, N=lane&15).
// ---------------------------------------------------------------------------
__global__ void qlstm_gemm(const float* __restrict__ X,
                           const float* __restrict__ Bmat,
                           const float* __restrict__ b16,
                           float* __restrict__ Zx) {
    int wave = (blockIdx.x * blockDim.x + threadIdx.x) >> 5;
    int lane = threadIdx.x & 31;
    int p    = lane & 15;
    int koff = (lane < 16) ? 0 : 2;
    long row0 = (long)wave * 16;
    const float* xrow = X + (row0 + p) * Dn + koff;     // A operand rows
    const float* bcol = Bmat + p;                        // B operand column

    v8f acc = {};
    #pragma unroll 4
    for (int k0 = 0; k0 < Dn; k0 += 4) {
        v2f a = *(const v2f*)(xrow + k0);               // K=k0+koff, k0+koff+1
        v2f b;
        b.x = bcol[(k0 + koff) * NP];
        b.y = bcol[(k0 + koff + 1) * NP];
        acc = __builtin_amdgcn_wmma_f32_16x16x4_f32(
            /*neg_a=*/false, a, /*neg_b=*/false, b,
            /*c_mod=*/(short)0, acc, /*reuse_a=*/false, /*reuse_b=*/false);
    }

    float bias  = b16[p];
    long  mbase = row0 + ((lane < 16) ? 0 : 8);
    #pragma unroll
    for (int r = 0; r < 8; ++r)
        Zx[(mbase + r) * NP + p] = acc[r] + bias;
}

// ---------------------------------------------------------------------------
// Kernel 2: per-batch scalar recurrence (c and h are uniform across H).
// angles[p] = Zx[t,b,p] + h * ws16[p];  <Z0> closed form; gate nonlinearities.
// ---------------------------------------------------------------------------
__global__ void qlstm_scan(const float* __restrict__ Zx,
                           const float* __restrict__ ws16,
                           float* __restrict__ Hs, float* __restrict__ Cfin) {
    int b = blockIdx.x * 32 + threadIdx.x;
    float wsv[NP];
    #pragma unroll
    for (int pp = 0; pp < NP; ++pp) wsv[pp] = ws16[pp];

    float h = 0.0f, c = 0.0f;
    for (int t = 0; t < Tn; ++t) {
        const float* z = Zx + ((long)t * Bn + b) * NP;
        v4f z0 = *(const v4f*)(z + 0);
        v4f z1 = *(const v4f*)(z + 4);
        v4f z2 = *(const v4f*)(z + 8);
        v4f z3 = *(const v4f*)(z + 12);
        float ang[NP];
        #pragma unroll
        for (int k = 0; k < 4; ++k) {
            ang[0 + k]  = z0[k] + h * wsv[0 + k];
            ang[4 + k]  = z1[k] + h * wsv[4 + k];
            ang[8 + k]  = z2[k] + h * wsv[8 + k];
            ang[12 + k] = z3[k] + h * wsv[12 + k];
        }
        float e[4];
        #pragma unroll
        for (int g = 0; g < 4; ++g) {
            float a0 = ang[4 * g + 0], a1 = ang[4 * g + 1], a2 = ang[4 * g + 2];
            e[g] = __cosf(a0) * __cosf(a2) - __sinf(a0) * __sinf(a1) * __sinf(a2);
        }
        float fg = fast_sigmoid(e[0]);
        float ig = fast_sigmoid(e[1]);
        float ug = fast_tanh(e[2]);
        float og = fast_sigmoid(e[3]);
        c = fg * c + ig * ug;
        h = og * fast_tanh(c);
        Hs[(long)t * Bn + b] = h;
    }
    Cfin[b] = c;
}

// ---------------------------------------------------------------------------
// Kernel 3: broadcast h scalars over H into outputs / hT / cT (~258 MB).
// One block per output row; 128 threads x float4 = 512 floats.
// ---------------------------------------------------------------------------
__global__ void qlstm_bcast(const float* __restrict__ Hs,
                            const float* __restrict__ Cfin,
                            float* __restrict__ out) {
    const long TB = (long)Tn * Bn;
    long bid = blockIdx.x;
    float v;
    float* dst;
    if (bid < TB) {                      // outputs[t,b,:]
        v = Hs[bid];
        dst = out + bid * Hn;
    } else if (bid < TB + Bn) {          // hT[b,:]
        long b = bid - TB;
        v = Hs[(long)(Tn - 1) * Bn + b];
        dst = out + TB * Hn + b * Hn;
    } else {                             // cT[b,:]
        long b = bid - TB - Bn;
        v = Cfin[b];
        dst = out + TB * Hn + (long)Bn * Hn + b * Hn;
    }
    v4f vv = {v, v, v, v};
    *(v4f*)(dst + (long)threadIdx.x * 4) = vv;
}

extern "C" void kernel_launch(void* const* d_in, const int* in_sizes, int n_in,
                              void* d_out, int out_size, void* d_ws, size_t ws_size,
                              hipStream_t stream) {
    const float* X  = (const float*)d_in[0];
    const float* Wf = (const float*)d_in[3];
    const float* bf = (const float*)d_in[4];
    const float* Wi = (const float*)d_in[5];
    const float* bi = (const float*)d_in[6];
    const float* Wu = (const float*)d_in[7];
    const float* bu = (const float*)d_in[8];
    const float* Wo = (const float*)d_in[9];
    const float* bo = (const float*)d_in[10];
    float* out = (float*)d_out;

    float* wsf  = (float*)d_ws;
    float* Bmat = wsf;                              // 512*16 = 8192 floats
    float* b16  = wsf + 8192;                       // 16
    float* ws16 = wsf + 8208;                       // 16
    float* Zx   = wsf + 8704;                       // T*B*16 = 2,097,152 (64B aligned)
    float* Hs   = Zx + (size_t)Tn * Bn * NP;        // T*B = 131,072
    float* Cfin = Hs + (size_t)Tn * Bn;             // B = 256
    (void)ws_size; (void)in_sizes; (void)n_in; (void)out_size;

    qlstm_pack<<<1, 512, 0, stream>>>(Wf, bf, Wi, bi, Wu, bu, Wo, bo, Bmat, b16, ws16);

    // 131072 rows / 16 rows-per-wave = 8192 waves; 8 waves (256 thr) per block
    qlstm_gemm<<<1024, 256, 0, stream>>>(X, Bmat, b16, Zx);

    qlstm_scan<<<Bn / 32, 32, 0, stream>>>(Zx, ws16, Hs, Cfin);

    qlstm_bcast<<<Tn * Bn + 2 * Bn, 128, 0, stream>>>(Hs, Cfin, out);
}